// BiDAStereo_78477642432747
// MI455X (gfx1250) — compile-verified
//
#include <hip/hip_runtime.h>

// Problem constants (from reference setup_inputs)
#define RATE  4
#define N_    4
#define C_    64
#define H_    96
#define W_    192
#define HR_   (H_ * RATE)   // 384
#define WR_   (W_ * RATE)   // 768
#define CCHUNK 16           // channels per block (grid.z = 64/16 = 4)

typedef float f32x4 __attribute__((ext_vector_type(4)));  // native vector for NT store

// ---- CDNA5 async copy: global -> LDS (per-lane B32), tracked by ASYNCcnt ----
__device__ __forceinline__ void async_ld_b32(unsigned lds_byte_off,
                                             unsigned long long gaddr) {
  asm volatile("global_load_async_to_lds_b32 %0, %1, off"
               :: "v"(lds_byte_off), "v"(gaddr)
               : "memory");
}

__device__ __forceinline__ void wait_async0() {
#if __has_builtin(__builtin_amdgcn_s_wait_asynccnt)
  __builtin_amdgcn_s_wait_asynccnt(0);
#else
  asm volatile("s_wait_asynccnt 0" ::: "memory");
#endif
}

// One block: one hi-res row (hy), one image n, 16 channels.
// Thread t = low-res column w; owns hi-res pixels hx = 4w..4w+3.
__global__ __launch_bounds__(W_) void fused_convex_upsample_warp(
    const float* __restrict__ x, const float* __restrict__ flow,
    const float* __restrict__ mask, float* __restrict__ out)
{
  // rows: comp*3 + di (di = -1,0,+1) ; col = w+1 ; cols 0 and 193 stay zero (padding)
  __shared__ float sflow[6][W_ + 8];

  const int w  = threadIdx.x;     // low-res column 0..191
  const int hy = blockIdx.x;      // hi-res row 0..383
  const int n  = blockIdx.y;      // image
  const int c0 = blockIdx.z * CCHUNK;
  const int h  = hy >> 2;
  const int ry = hy & 3;

  // ---- zero-init LDS (gives the zero padding of jnp.pad for free) ----
  {
    float* s = &sflow[0][0];
    for (int i = w; i < 6 * (W_ + 8); i += W_) s[i] = 0.0f;
  }
  __syncthreads();

  // ---- async-stage 3 flow rows x 2 components into LDS (coalesced) ----
  #pragma unroll
  for (int comp = 0; comp < 2; ++comp) {
    #pragma unroll
    for (int di = 0; di < 3; ++di) {
      const int hh = h + di - 1;
      if (hh >= 0 && hh < H_) {   // block-uniform branch -> scalar skip
        const float* g = flow + ((size_t)((n * 2 + comp) * H_ + hh)) * W_ + w;
        const unsigned lds = (unsigned)(unsigned long long)&sflow[comp * 3 + di][w + 1];
        async_ld_b32(lds, (unsigned long long)g);
      }
    }
  }
  wait_async0();
  __syncthreads();

  // ---- 3x3 taps of (rate * flow), zero-padded ----
  float fx[9], fy[9];
  #pragma unroll
  for (int di = 0; di < 3; ++di)
    #pragma unroll
    for (int dj = 0; dj < 3; ++dj) {
      fx[di * 3 + dj] = (float)RATE * sflow[di][w + dj];
      fy[di * 3 + dj] = (float)RATE * sflow[3 + di][w + dj];
    }

  // ---- per-pixel softmax + convex combo + bilinear setup (4 pixels/thread) ----
  int   gidx[4][4];   // gather offsets within one channel plane
  float gwt [4][4];   // validity-masked bilinear weights

  #pragma unroll
  for (int rx = 0; rx < 4; ++rx) {
    float m[9];
    #pragma unroll
    for (int k = 0; k < 9; ++k)   // mask streamed once -> non-temporal
      m[k] = __builtin_nontemporal_load(
          mask + ((size_t)(n * 144 + k * 16 + ry * 4 + rx) * H_ + h) * W_ + w);

    float mx = m[0];
    #pragma unroll
    for (int k = 1; k < 9; ++k) mx = fmaxf(mx, m[k]);

    float s = 0.f, ufx = 0.f, ufy = 0.f;
    #pragma unroll
    for (int k = 0; k < 9; ++k) {
      const float e = __expf(m[k] - mx);
      s += e;
      ufx = fmaf(e, fx[k], ufx);
      ufy = fmaf(e, fy[k], ufy);
    }
    const float inv = 1.0f / s;
    ufx *= inv; ufy *= inv;

    const int   hx = (w << 2) + rx;
    const float sx = (float)hx + ufx;
    const float sy = (float)hy + ufy;
    const float x0f = floorf(sx), y0f = floorf(sy);
    const float wx1 = sx - x0f, wx0 = 1.f - wx1;
    const float wy1 = sy - y0f, wy0 = 1.f - wy1;

    const float xs[2]  = {x0f, x0f + 1.f};
    const float ys[2]  = {y0f, y0f + 1.f};
    const float wxs[2] = {wx0, wx1};
    const float wys[2] = {wy0, wy1};
    #pragma unroll
    for (int cy = 0; cy < 2; ++cy)
      #pragma unroll
      for (int cx = 0; cx < 2; ++cx) {
        const float xf = xs[cx], yf = ys[cy];
        const bool v = (xf >= 0.f) && (xf <= (float)(WR_ - 1)) &&
                       (yf >= 0.f) && (yf <= (float)(HR_ - 1));
        const int xc = (int)fminf(fmaxf(xf, 0.f), (float)(WR_ - 1));
        const int yc = (int)fminf(fmaxf(yf, 0.f), (float)(HR_ - 1));
        gidx[rx][cy * 2 + cx] = yc * WR_ + xc;
        gwt [rx][cy * 2 + cx] = v ? wxs[cx] * wys[cy] : 0.0f;
      }
  }

  // ---- channel loop: 16 gathers + blend + one aligned 16B NT store ----
  const size_t plane = (size_t)HR_ * WR_;
  const float* xp = x   + (size_t)(n * C_ + c0) * plane;
  float*       op = out + (size_t)(n * C_ + c0) * plane + (size_t)hy * WR_ + (w << 2);

  #pragma unroll 2
  for (int c = 0; c < CCHUNK; ++c) {
    if (c + 1 < CCHUNK) {   // prefetch next channel plane's gather lines
      const float* xn = xp + plane;
      __builtin_prefetch(xn + gidx[0][0], 0, 1);
      __builtin_prefetch(xn + gidx[1][3], 0, 1);
      __builtin_prefetch(xn + gidx[2][0], 0, 1);
      __builtin_prefetch(xn + gidx[3][3], 0, 1);
    }
    f32x4 o;
    #pragma unroll
    for (int rx = 0; rx < 4; ++rx) {
      float a = 0.f;
      #pragma unroll
      for (int j = 0; j < 4; ++j)
        a = fmaf(gwt[rx][j], xp[gidx[rx][j]], a);
      o[rx] = a;
    }
    __builtin_nontemporal_store(o, reinterpret_cast<f32x4*>(op));
    xp += plane;
    op += plane;
  }
}

extern "C" void kernel_launch(void* const* d_in, const int* in_sizes, int n_in,
                              void* d_out, int out_size, void* d_ws, size_t ws_size,
                              hipStream_t stream) {
  const float* x    = (const float*)d_in[0];   // [4,64,384,768]
  const float* flow = (const float*)d_in[1];   // [4,2,96,192]
  const float* mask = (const float*)d_in[2];   // [4,144,96,192]
  float* out = (float*)d_out;                  // [4,64,384,768]

  dim3 grid(HR_, N_, C_ / CCHUNK);   // 384 x 4 x 4 = 6144 blocks
  dim3 block(W_, 1, 1);              // 192 threads = 6 wave32
  fused_convex_upsample_warp<<<grid, block, 0, stream>>>(x, flow, mask, out);

  (void)in_sizes; (void)n_in; (void)out_size; (void)d_ws; (void)ws_size;
}